// Model_53893249630756
// MI455X (gfx1250) — compile-verified
//
#include <hip/hip_runtime.h>
#include <math.h>

typedef __attribute__((ext_vector_type(2))) float v2f;
typedef __attribute__((ext_vector_type(8))) float v8f;

#define N_NODES 50000
#define N_REL 8
#define D_IN 64
#define HF 64          // H * D_OUT
#define NHEAD 2
#define D_OUT 32
#define NEG_SLOPE 0.2f

// ---------------------------------------------------------------------------
// helpers
// ---------------------------------------------------------------------------
__device__ __forceinline__ void atomicMaxFloat(float* addr, float val) {
    // works for any sign given init to -inf:
    //  positive values: int-compare is order-preserving
    //  negative values: uint-compare is order-reversing -> atomicMin
    if (val >= 0.0f) {
        atomicMax((int*)addr, __float_as_int(val));
    } else {
        atomicMin((unsigned int*)addr, __float_as_uint(val));
    }
}

__device__ __forceinline__ void atomAddF(float* p, float v) {
    unsafeAtomicAdd(p, v);   // native global_atomic_add_f32 on CDNA
}

// ---------------------------------------------------------------------------
// 0) init: zero output + denom, set segment-max buffer to -inf
// ---------------------------------------------------------------------------
__global__ void init_kernel(float* __restrict__ out, int out_n,
                            float* __restrict__ emax,
                            float* __restrict__ denom, int nh) {
    int i = blockIdx.x * blockDim.x + threadIdx.x;
    if (i < out_n) out[i] = 0.0f;
    if (i < nh) {
        emax[i]  = -__builtin_huge_valf();
        denom[i] = 0.0f;
    }
}

// ---------------------------------------------------------------------------
// 1) dense per-(node, relation) projection with V_WMMA_F32_16X16X4_F32
//    P[n, r, :] = X[n, :] @ W[r] + B[r]      (X: [N,64], W: [R,64,64])
//    one wave handles a 16-node tile for one relation (C = 16x64)
// ---------------------------------------------------------------------------
__global__ __launch_bounds__(256) void proj_wmma_kernel(
    const float* __restrict__ X,      // [N_NODES, D_IN]
    const float* __restrict__ W,      // [N_REL, D_IN, HF]
    const float* __restrict__ Bv,     // [N_REL, HF]
    float* __restrict__ P)            // [N_NODES, N_REL, HF]
{
    const int lane   = threadIdx.x & 31;
    const int gw     = blockIdx.x * 8 + (threadIdx.x >> 5);   // global wave id
    const int ntiles = (N_NODES + 15) / 16;                   // 3125
    if (gw >= ntiles * N_REL) return;                         // wave-uniform

    const int r  = gw / ntiles;
    const int n0 = (gw % ntiles) * 16;

    const int  row   = lane & 15;          // M within tile
    const int  khalf = (lane >> 4) << 1;   // 0 (lanes 0-15) or 2 (lanes 16-31)
    const int  node  = n0 + row;
    const bool valid = node < N_NODES;

    // A fragments for all 16 K-steps (ISA 16x4 f32 layout)
    v2f afrag[16];
    const float* xrow = X + (size_t)node * D_IN;
#pragma unroll
    for (int ks = 0; ks < 16; ++ks) {
        const int k = ks * 4 + khalf;
        afrag[ks].x = valid ? xrow[k]     : 0.0f;
        afrag[ks].y = valid ? xrow[k + 1] : 0.0f;
    }

    const float* Wr = W + (size_t)r * D_IN * HF;

#pragma unroll
    for (int c = 0; c < 4; ++c) {                 // 4 column tiles of 16
        const int col = c * 16 + (lane & 15);     // N index this lane covers
        v8f acc = {};
#pragma unroll
        for (int ks = 0; ks < 16; ++ks) {
            const int k = ks * 4 + khalf;
            v2f bfrag;
            bfrag.x = Wr[(size_t)k       * HF + col];
            bfrag.y = Wr[(size_t)(k + 1) * HF + col];
            acc = __builtin_amdgcn_wmma_f32_16x16x4_f32(
                false, afrag[ks], false, bfrag, (short)0, acc, false, false);
        }
        const float bias  = Bv[r * HF + col];
        const int   mbase = (lane < 16) ? 0 : 8;
#pragma unroll
        for (int i = 0; i < 8; ++i) {
            const int n = n0 + mbase + i;
            if (n < N_NODES)
                P[((size_t)n * N_REL + r) * HF + col] = acc[i] + bias;
        }
    }
}

// ---------------------------------------------------------------------------
// 2) per-edge attention logits + segment max (1 wave per edge)
// ---------------------------------------------------------------------------
__global__ __launch_bounds__(256) void edge_logit_kernel(
    const float* __restrict__ Psrc, const float* __restrict__ Pqual,
    const int* __restrict__ src, const int* __restrict__ dst,
    const int* __restrict__ rt,  const int* __restrict__ nid,
    const float* __restrict__ attn,   // [N_REL, NHEAD, D_OUT]
    float* __restrict__ elog,         // [E, 2]
    float* __restrict__ emax,         // [N_NODES, 2]
    int E)
{
    const int e = blockIdx.x * 8 + (threadIdx.x >> 5);
    if (e >= E) return;
    const int lane = threadIdx.x & 31;

    const int s = src[e], q = nid[e], r = rt[e], d = dst[e];
    const float* ps = Psrc  + ((size_t)s * N_REL + r) * HF;
    const float* pq = Pqual + ((size_t)q * N_REL + r) * HF;

    float x0 = ps[lane]      + pq[lane];         // head 0 dims 0..31
    float x1 = ps[lane + 32] + pq[lane + 32];    // head 1 dims 32..63
    x0 = (x0 >= 0.0f) ? x0 : x0 * NEG_SLOPE;
    x1 = (x1 >= 0.0f) ? x1 : x1 * NEG_SLOPE;

    const float* ar = attn + r * (NHEAD * D_OUT);
    float e0 = x0 * ar[lane];
    float e1 = x1 * ar[32 + lane];

#pragma unroll
    for (int off = 16; off > 0; off >>= 1) {
        e0 += __shfl_xor(e0, off, 32);
        e1 += __shfl_xor(e1, off, 32);
    }
    if (lane == 0) {
        elog[2 * e]     = e0;
        elog[2 * e + 1] = e1;
        atomicMaxFloat(&emax[2 * d],     e0);
        atomicMaxFloat(&emax[2 * d + 1], e1);
    }
}

// ---------------------------------------------------------------------------
// 3) exp(e - emax[dst]) + segment sum (one thread per (edge, head))
// ---------------------------------------------------------------------------
__global__ void edge_exp_kernel(const int* __restrict__ dst,
                                const float* __restrict__ emax,
                                float* __restrict__ elog,
                                float* __restrict__ denom, int E2) {
    const int i = blockIdx.x * blockDim.x + threadIdx.x;
    if (i >= E2) return;
    const int e = i >> 1, h = i & 1;
    const int d  = dst[e];
    const float ex = __expf(elog[i] - emax[2 * d + h]);
    elog[i] = ex;
    atomAddF(&denom[2 * d + h], ex);
}

// ---------------------------------------------------------------------------
// 4) normalized attention * raw source features -> scatter into rst
// ---------------------------------------------------------------------------
__global__ __launch_bounds__(256) void edge_msg_kernel(
    const float* __restrict__ feat, const int* __restrict__ src,
    const int* __restrict__ dst,
    const float* __restrict__ elog, const float* __restrict__ denom,
    float* __restrict__ out, int E)
{
    const int e = blockIdx.x * 8 + (threadIdx.x >> 5);
    if (e >= E) return;
    const int lane = threadIdx.x & 31;
    const int s = src[e], d = dst[e];

    const float a0 = elog[2 * e]     / denom[2 * d];
    const float a1 = elog[2 * e + 1] / denom[2 * d + 1];

    const float f0 = feat[(size_t)s * D_IN + lane];
    const float f1 = feat[(size_t)s * D_IN + lane + 32];

    float* o = out + (size_t)d * (NHEAD * D_IN);
    atomAddF(o + lane,       f0 * a0);
    atomAddF(o + 32 + lane,  f1 * a0);
    atomAddF(o + 64 + lane,  f0 * a1);
    atomAddF(o + 96 + lane,  f1 * a1);
}

// ---------------------------------------------------------------------------
// launch
// ---------------------------------------------------------------------------
extern "C" void kernel_launch(void* const* d_in, const int* in_sizes, int n_in,
                              void* d_out, int out_size, void* d_ws, size_t ws_size,
                              hipStream_t stream) {
    const float* feat      = (const float*)d_in[0];
    const int*   src       = (const int*)  d_in[1];
    const int*   dst       = (const int*)  d_in[2];
    const int*   r_type    = (const int*)  d_in[3];
    const int*   nid       = (const int*)  d_in[4];
    const float* fc_src    = (const float*)d_in[5];
    const float* fc_src_b  = (const float*)d_in[6];
    const float* node_qual = (const float*)d_in[7];
    const float* fc_qual   = (const float*)d_in[8];
    const float* fc_qual_b = (const float*)d_in[9];
    const float* attn      = (const float*)d_in[10];

    const int E = in_sizes[1];
    float* out = (float*)d_out;

    // workspace layout (floats)
    const size_t NP = (size_t)N_NODES * N_REL * HF;   // 25.6M per projection
    float* ws    = (float*)d_ws;
    float* Psrc  = ws;
    float* Pqual = ws + NP;
    float* elog  = ws + 2 * NP;                       // E*2
    float* emax  = elog + (size_t)E * 2;              // N*2
    float* denom = emax + (size_t)N_NODES * 2;        // N*2

    // 0) init
    {
        const int out_n = N_NODES * NHEAD * D_IN;
        const int nh    = N_NODES * NHEAD;
        const int n     = out_n;                       // out_n > nh
        init_kernel<<<(n + 255) / 256, 256, 0, stream>>>(out, out_n, emax, denom, nh);
    }
    // 1) projections (WMMA)
    {
        const int ntiles = (N_NODES + 15) / 16;        // 3125
        const int waves  = ntiles * N_REL;             // 25000
        const int blocks = (waves + 7) / 8;            // 3125
        proj_wmma_kernel<<<blocks, 256, 0, stream>>>(feat,      fc_src,  fc_src_b,  Psrc);
        proj_wmma_kernel<<<blocks, 256, 0, stream>>>(node_qual, fc_qual, fc_qual_b, Pqual);
    }
    // 2) logits + segment max
    edge_logit_kernel<<<(E + 7) / 8, 256, 0, stream>>>(
        Psrc, Pqual, src, dst, r_type, nid, attn, elog, emax, E);
    // 3) exp + segment sum
    edge_exp_kernel<<<(2 * E + 255) / 256, 256, 0, stream>>>(dst, emax, elog, denom, 2 * E);
    // 4) weighted message scatter
    edge_msg_kernel<<<(E + 7) / 8, 256, 0, stream>>>(feat, src, dst, elog, denom, out, E);
}